// RelativeTransformerBlock_6914897347238
// MI455X (gfx1250) — compile-verified
//
#include <hip/hip_runtime.h>
#include <hip/hip_bf16.h>
#include <math.h>
#include <stdint.h>

// Shapes (fixed by the reference)
#define BB 2
#define LL 512
#define CC 256
#define HH 8
#define DD 32
#define MLPN 1024   // RATIO*C
#define NROWS (BB*LL)            // 1024
#define BH (BB*HH)               // 16
#define LSQ (LL*LL)              // 262144
#define EPSV 1e-5f
#define SCALEV 0.17677669529663687f  // 1/sqrt(32)

typedef __attribute__((ext_vector_type(2))) float v2f;
typedef __attribute__((ext_vector_type(8))) float v8f;

__device__ __forceinline__ float gelu_exact(float x) {
    return 0.5f * x * (1.0f + erff(x * 0.70710678118654752f));
}

// One wave computes a 16x16 f32 tile of A[M,K] @ B[K,N] using V_WMMA_F32_16X16X4_F32.
// Fragment layout (wave32):
//   A: lane l holds A[l&15][k + 2*(l>>4)] , A[l&15][k + 2*(l>>4) + 1]
//   B: lane l holds B[k + 2*(l>>4)][l&15] , B[k + 2*(l>>4)+1][l&15]
//   C: lane l, vgpr r -> C[r + 8*(l>>4)][l&15]
__device__ __forceinline__ v8f wmma_gemm_f32(const float* __restrict__ A, int lda,
                                             const float* __restrict__ B, int ldb, int K) {
    const unsigned lane = threadIdx.x & 31u;
    const unsigned n    = lane & 15u;
    const unsigned kh   = lane >> 4;       // 0 or 1
    const float* arow   = A + (lane & 15u) * lda;
    v8f acc = {};
    for (int k = 0; k < K; k += 4) {
        v2f a = *reinterpret_cast<const v2f*>(arow + k + 2 * kh);
        v2f b;
        b.x = B[(k + 2 * kh) * ldb + n];
        b.y = B[(k + 2 * kh + 1) * ldb + n];
        acc = __builtin_amdgcn_wmma_f32_16x16x4_f32(false, a, false, b,
                                                    (short)0, acc, false, false);
    }
    return acc;
}

// ---------------- LayerNorm over C=256 (one block per row) ----------------
__global__ void ln_kernel(const float* __restrict__ x, const float* __restrict__ g,
                          const float* __restrict__ b, float* __restrict__ out) {
    const int row = blockIdx.x;
    const int c   = threadIdx.x;   // 256 threads
    __shared__ float red[CC];
    float v = x[row * CC + c];
    red[c] = v;
    __syncthreads();
    for (int s = 128; s > 0; s >>= 1) { if (c < s) red[c] += red[c + s]; __syncthreads(); }
    float mean = red[0] * (1.0f / CC);
    __syncthreads();
    float d = v - mean;
    red[c] = d * d;
    __syncthreads();
    for (int s = 128; s > 0; s >>= 1) { if (c < s) red[c] += red[c + s]; __syncthreads(); }
    float var = red[0] * (1.0f / CC);
    out[row * CC + c] = d * rsqrtf(var + EPSV) * g[c] + b[c];
}

// ---------------- qkv = h @ w_qkv  (1024x768x256), scatter to q/kT/v ----------------
__global__ void qkv_kernel(const float* __restrict__ h, const float* __restrict__ w,
                           float* __restrict__ qb, float* __restrict__ ktb,
                           float* __restrict__ vb) {
    const int t  = blockIdx.x * 4 + (threadIdx.x >> 5);   // 3072 tiles
    const int mt = t % 64, nt = t / 64;                   // 64 x 48 tiles
    v8f acc = wmma_gemm_f32(h + mt * 16 * CC, CC, w + nt * 16, 3 * CC, CC);
    const unsigned lane = threadIdx.x & 31u;
    const unsigned nl = lane & 15u, mh = lane >> 4;
    const int nabs  = nt * 16 + (int)nl;
    const int which = nabs / CC;             // 0=q 1=k 2=v (uniform per tile)
    const int hd    = (nabs % CC) / DD;      // head (uniform per tile)
    const int d     = nabs % DD;
    for (int r = 0; r < 8; ++r) {
        int m  = mt * 16 + r + 8 * (int)mh;
        int b  = m / LL, l = m % LL;
        int bh = b * HH + hd;
        float val = acc[r];
        if (which == 0)      qb [(bh * LL + l) * DD + d] = val;
        else if (which == 1) ktb[(bh * DD + d) * LL + l] = val;
        else                 vb [(bh * LL + l) * DD + d] = val;
    }
}

// ------- rel: stream relative_pos (512MB) via async global->LDS, WMMA project to 8 heads,
//         GELU, LayerNorm over heads, scatter to [B,H,L,L] -------
#define AROW_PAD 272   // 64 floats (256B) + 16B pad per row chunk
__global__ void rel_kernel(const float* __restrict__ rp, const float* __restrict__ w_rel,
                           const float* __restrict__ b_rel, const float* __restrict__ g_ln,
                           const float* __restrict__ b_ln, float* __restrict__ rel_out) {
    __shared__ float Bs[CC * 16];                       // w_rel padded 256x16 (cols 8..15 = 0)
    __shared__ float tileS[4][16][17];                  // per-wave 16x16 staging
    __shared__ __align__(16) char Abuf[4][2][16 * AROW_PAD];  // per-wave double buffer
    const int tid = threadIdx.x;                        // 128 threads = 4 waves
    for (int idx = tid; idx < CC * 16; idx += 128) {
        int k = idx >> 4, nn = idx & 15;
        Bs[idx] = (nn < HH) ? w_rel[k * HH + nn] : 0.0f;
    }
    __syncthreads();

    const int wave = tid >> 5;
    const unsigned lane = tid & 31u;
    const int t   = blockIdx.x * 4 + wave;     // 32768 tiles total
    const int j16 = t & 31;                    // L/16 = 32
    const int i   = (t >> 5) & 511;
    const int b   = t >> 14;

    const float* A = rp + ((size_t)((b * LL + i) * LL + j16 * 16)) * CC;
    const unsigned n  = lane & 15u;
    const unsigned kh = lane >> 4;

    // async staging lane map: row = lane>>1, 16-byte segment = (lane&1)*16
    const unsigned srow = lane >> 1;
    const unsigned sseg = (lane & 1u) * 16u;
    const uint64_t gbase = (uint64_t)(uintptr_t)((const char*)(A + srow * CC) + sseg);
    unsigned lbase0 = ((unsigned)(uintptr_t)(&Abuf[wave][0][0])) + srow * AROW_PAD + sseg;
    unsigned lbase1 = ((unsigned)(uintptr_t)(&Abuf[wave][1][0])) + srow * AROW_PAD + sseg;

    // prologue: chunk 0 -> buffer 0 (8 x b128 per wave = 4KB)
    {
        uint64_t ga = gbase;
        unsigned la = lbase0;
        #pragma unroll
        for (int u = 0; u < 8; ++u) {
            asm volatile("global_load_async_to_lds_b128 %0, %1, off"
                         :: "v"(la), "v"(ga) : "memory");
            la += 32u; ga += 32u;
        }
    }

    v8f acc = {};
    int buf = 0;
    for (int kc = 0; kc < 4; ++kc) {
        if (kc < 3) {  // prefetch next chunk into other buffer
            uint64_t ga = gbase + (uint64_t)(kc + 1) * 256u;
            unsigned la = (buf == 0) ? lbase1 : lbase0;
            #pragma unroll
            for (int u = 0; u < 8; ++u) {
                asm volatile("global_load_async_to_lds_b128 %0, %1, off"
                             :: "v"(la), "v"(ga) : "memory");
                la += 32u; ga += 32u;
            }
            // 8 new ops outstanding; <=8 means previous chunk fully landed (in-order)
            asm volatile("s_wait_asynccnt 0x8" ::: "memory");
        } else {
            asm volatile("s_wait_asynccnt 0x0" ::: "memory");
        }
        const char* abase = &Abuf[wave][buf][0];
        const float* rowp = (const float*)(abase + (lane & 15u) * AROW_PAD);
        #pragma unroll
        for (int kk = 0; kk < 64; kk += 4) {
            v2f a = *(const v2f*)(rowp + kk + 2 * kh);
            v2f bb;
            int kg = kc * 64 + kk + 2 * (int)kh;
            bb.x = Bs[kg * 16 + n];
            bb.y = Bs[(kg + 1) * 16 + n];
            acc = __builtin_amdgcn_wmma_f32_16x16x4_f32(false, a, false, bb,
                                                        (short)0, acc, false, false);
        }
        buf ^= 1;
    }

    for (int r = 0; r < 8; ++r) tileS[wave][r + 8 * (int)kh][n] = acc[r];
    __syncthreads();

    if (lane < 16) {
        float v[HH];
        float s = 0.0f;
        for (int hh = 0; hh < HH; ++hh) {
            float xg = gelu_exact(tileS[wave][lane][hh] + b_rel[hh]);
            v[hh] = xg;
            s += xg;
        }
        float mean = s * (1.0f / HH);
        float s2 = 0.0f;
        for (int hh = 0; hh < HH; ++hh) { float d = v[hh] - mean; s2 += d * d; }
        float inv = rsqrtf(s2 * (1.0f / HH) + EPSV);
        const int j = j16 * 16 + (int)lane;
        for (int hh = 0; hh < HH; ++hh) {
            rel_out[((b * HH + hh) * LL + i) * LL + j] =
                (v[hh] - mean) * inv * g_ln[hh] + b_ln[hh];
        }
    }
}

// ---------------- scores = Q @ K^T, fused (+rel)*scale ----------------
__global__ void scores_kernel(const float* __restrict__ q, const float* __restrict__ kt,
                              const float* __restrict__ rel, float* __restrict__ p) {
    const int t   = blockIdx.x * 4 + (threadIdx.x >> 5);  // 16384 tiles
    const int bh  = t >> 10;
    const int rem = t & 1023;
    const int mt  = rem & 31, nt = rem >> 5;
    v8f acc = wmma_gemm_f32(q + bh * LL * DD + mt * 16 * DD, DD,
                            kt + bh * DD * LL + nt * 16, LL, DD);
    const unsigned lane = threadIdx.x & 31u;
    const int j  = nt * 16 + (int)(lane & 15u);
    const int mh = (int)(lane >> 4);
    for (int r = 0; r < 8; ++r) {
        int i   = mt * 16 + r + 8 * mh;
        int idx = bh * LSQ + i * LL + j;
        p[idx] = (acc[r] + rel[idx]) * SCALEV;
    }
}

// ---------------- softmax over last dim (512), in place ----------------
__global__ void softmax_kernel(float* __restrict__ p) {
    float* pr = p + (size_t)blockIdx.x * LL;    // 8192 rows
    const int t = threadIdx.x;                  // 256 threads, 2 elems each
    __shared__ float red[256];
    float a = pr[t], b = pr[t + 256];
    red[t] = fmaxf(a, b);
    __syncthreads();
    for (int s = 128; s > 0; s >>= 1) { if (t < s) red[t] = fmaxf(red[t], red[t + s]); __syncthreads(); }
    float mx = red[0];
    __syncthreads();
    float ea = __expf(a - mx), eb = __expf(b - mx);
    red[t] = ea + eb;
    __syncthreads();
    for (int s = 128; s > 0; s >>= 1) { if (t < s) red[t] += red[t + s]; __syncthreads(); }
    float inv = 1.0f / red[0];
    pr[t] = ea * inv;
    pr[t + 256] = eb * inv;
}

// ---------------- O = P @ V, scatter to [B,L,C] row layout ----------------
__global__ void av_kernel(const float* __restrict__ p, const float* __restrict__ v,
                          float* __restrict__ orow) {
    const int t   = blockIdx.x * 4 + (threadIdx.x >> 5);  // 1024 tiles
    const int bh  = t >> 6;
    const int rem = t & 63;
    const int mt  = rem >> 1, nt = rem & 1;
    v8f acc = wmma_gemm_f32(p + bh * LSQ + mt * 16 * LL, LL,
                            v + bh * LL * DD + nt * 16, DD, LL);
    const unsigned lane = threadIdx.x & 31u;
    const int d  = nt * 16 + (int)(lane & 15u);
    const int mh = (int)(lane >> 4);
    const int hh = bh & 7, b = bh >> 3;
    for (int r = 0; r < 8; ++r) {
        int l = mt * 16 + r + 8 * mh;
        orow[(b * LL + l) * CC + hh * DD + d] = acc[r];
    }
}

// ---------------- x1 = o @ w_proj + b_proj + x ----------------
__global__ void proj_kernel(const float* __restrict__ o, const float* __restrict__ w,
                            const float* __restrict__ bias, const float* __restrict__ x,
                            float* __restrict__ x1) {
    const int t  = blockIdx.x * 4 + (threadIdx.x >> 5);   // 1024 tiles
    const int mt = t & 63, nt = t >> 6;
    v8f acc = wmma_gemm_f32(o + mt * 16 * CC, CC, w + nt * 16, CC, CC);
    const unsigned lane = threadIdx.x & 31u;
    const int nc = nt * 16 + (int)(lane & 15u);
    const int mh = (int)(lane >> 4);
    for (int r = 0; r < 8; ++r) {
        int m = mt * 16 + r + 8 * mh;
        x1[m * CC + nc] = acc[r] + bias[nc] + x[m * CC + nc];
    }
}

// ---------------- mid = GELU(h2 @ w_mlp1 + b_mlp1) ----------------
__global__ void mlp1_kernel(const float* __restrict__ h, const float* __restrict__ w,
                            const float* __restrict__ bias, float* __restrict__ mid) {
    const int t  = blockIdx.x * 4 + (threadIdx.x >> 5);   // 4096 tiles
    const int mt = t & 63, nt = t >> 6;
    v8f acc = wmma_gemm_f32(h + mt * 16 * CC, CC, w + nt * 16, MLPN, CC);
    const unsigned lane = threadIdx.x & 31u;
    const int nc = nt * 16 + (int)(lane & 15u);
    const int mh = (int)(lane >> 4);
    for (int r = 0; r < 8; ++r) {
        int m = mt * 16 + r + 8 * mh;
        mid[m * MLPN + nc] = gelu_exact(acc[r] + bias[nc]);
    }
}

// ---------------- out = mid @ w_mlp2 + b_mlp2 + x1 ----------------
__global__ void mlp2_kernel(const float* __restrict__ mid, const float* __restrict__ w,
                            const float* __restrict__ bias, const float* __restrict__ x1,
                            float* __restrict__ out) {
    const int t  = blockIdx.x * 4 + (threadIdx.x >> 5);   // 1024 tiles
    const int mt = t & 63, nt = t >> 6;
    v8f acc = wmma_gemm_f32(mid + mt * 16 * MLPN, MLPN, w + nt * 16, CC, MLPN);
    const unsigned lane = threadIdx.x & 31u;
    const int nc = nt * 16 + (int)(lane & 15u);
    const int mh = (int)(lane >> 4);
    for (int r = 0; r < 8; ++r) {
        int m = mt * 16 + r + 8 * mh;
        out[m * CC + nc] = acc[r] + bias[nc] + x1[m * CC + nc];
    }
}

extern "C" void kernel_launch(void* const* d_in, const int* in_sizes, int n_in,
                              void* d_out, int out_size, void* d_ws, size_t ws_size,
                              hipStream_t stream) {
    const float* x       = (const float*)d_in[0];
    const float* rp      = (const float*)d_in[1];
    const float* w_qkv   = (const float*)d_in[2];
    const float* w_proj  = (const float*)d_in[3];
    const float* b_proj  = (const float*)d_in[4];
    const float* w_rel   = (const float*)d_in[5];
    const float* b_rel   = (const float*)d_in[6];
    const float* g_reln  = (const float*)d_in[7];
    const float* b_reln  = (const float*)d_in[8];
    const float* g_n1    = (const float*)d_in[9];
    const float* b_n1    = (const float*)d_in[10];
    const float* g_n2    = (const float*)d_in[11];
    const float* b_n2    = (const float*)d_in[12];
    const float* w_mlp1  = (const float*)d_in[13];
    const float* b_mlp1  = (const float*)d_in[14];
    const float* w_mlp2  = (const float*)d_in[15];
    const float* b_mlp2  = (const float*)d_in[16];
    float* out = (float*)d_out;

    float* ws = (float*)d_ws;
    float* h1    = ws;                       // 1024*256
    float* qb    = h1    + NROWS * CC;       // 16*512*32
    float* ktb   = qb    + BH * LL * DD;
    float* vb    = ktb   + BH * DD * LL;
    float* relb  = vb    + BH * LL * DD;     // 16*512*512
    float* pbuf  = relb  + BH * LSQ;         // 16*512*512
    float* orow  = pbuf  + BH * LSQ;         // 1024*256
    float* x1    = orow  + NROWS * CC;       // 1024*256
    float* h2    = x1    + NROWS * CC;       // 1024*256
    float* mid   = h2    + NROWS * CC;       // 1024*1024

    ln_kernel     <<<NROWS, 256, 0, stream>>>(x, g_n1, b_n1, h1);
    qkv_kernel    <<<768,   128, 0, stream>>>(h1, w_qkv, qb, ktb, vb);
    rel_kernel    <<<8192,  128, 0, stream>>>(rp, w_rel, b_rel, g_reln, b_reln, relb);
    scores_kernel <<<4096,  128, 0, stream>>>(qb, ktb, relb, pbuf);
    softmax_kernel<<<BH * LL, 256, 0, stream>>>(pbuf);
    av_kernel     <<<256,   128, 0, stream>>>(pbuf, vb, orow);
    proj_kernel   <<<256,   128, 0, stream>>>(orow, w_proj, b_proj, x, x1);
    ln_kernel     <<<NROWS, 256, 0, stream>>>(x1, g_n2, b_n2, h2);
    mlp1_kernel   <<<1024,  128, 0, stream>>>(h2, w_mlp1, b_mlp1, mid);
    mlp2_kernel   <<<256,   128, 0, stream>>>(mid, w_mlp2, b_mlp2, x1, out);
}